// MCPBRNN_Generic_variable_Out_constantLoss_53695681134738
// MI455X (gfx1250) — compile-verified
//
#include <hip/hip_runtime.h>
#include <math.h>

// ---------------------------------------------------------------------------
// MCPBRNN scalar recurrence on MI455X (gfx1250, wave32).
//
// Plan:
//  kernel 1 (scan):  contraction-based block-parallel scan. 512 lanes, each
//                    owns a 1024-step chunk and re-runs a 1024-step warm-up
//                    (f < 1 strictly => initial-carry error < f^1024 ~ 0).
//  kernel 2 (std):   unbiased std of y_obs[1000:400000] via the matrix pipe:
//                    V_WMMA_F32_16X16X4_F32 with all-ones A gives 64-wide
//                    sums per issue; two accumulators give (sum, sumsq).
//  kernel 3 (fill):  constants / broadcast-std / time_lag zero padding.
// ---------------------------------------------------------------------------

typedef float v2f __attribute__((ext_vector_type(2)));
typedef float v8f __attribute__((ext_vector_type(8)));

#define SPIN_LEN      1000
#define TRAINTIME_LEN 400000
#define CHUNK         1024
#define WARM          1024

__device__ __forceinline__ float sigmoidf(float z) {
    return 1.0f / (1.0f + __expf(-z));
}

// ---------------------------------------------------------------------------
// Kernel 1: block-parallel nonlinear scan with warm-up windows.
// Output layout (flat, Bn = B):
//   [0,B)=h_n [B,2B)=c_n [2B,3B)=l_n [3B,4B)=bp_n [4B,5B)=Gate_ib
//   [5B,6B)=Gate_oo [6B,7B)=Gate_ol [7B,8B)=Gate_f [8B,10B)=h_nout[B,2]
//   [10B,11B)=obs_std
// ---------------------------------------------------------------------------
__global__ void mcpbrnn_scan_kernel(const float* __restrict__ x,
                                    const int*   __restrict__ time_lag_p,
                                    const float* __restrict__ c_mean,
                                    const float* __restrict__ c_stdp,
                                    const float* __restrict__ wo,
                                    const float* __restrict__ wl,
                                    const float* __restrict__ wf,
                                    const float* __restrict__ b0p,
                                    const float* __restrict__ b1p,
                                    float* __restrict__ out, int Bn) {
    const int tid = blockIdx.x * blockDim.x + threadIdx.x;
    const int tl  = time_lag_p[0];
    const long long N = (long long)Bn - tl;
    if (N <= 0) return;
    long long start = (long long)tid * CHUNK;
    if (start >= N) return;
    long long end = start + CHUNK;
    if (end > N) end = N;

    const float eo  = __expf(wo[0]);
    const float el  = __expf(wl[0]);
    const float ef  = __expf(wf[0]);
    const float den = eo + el + ef;
    const float oo1 = eo / den;
    const float ol  = el / den;
    const float mo  = c_mean[0];
    const float so  = c_stdp[0];
    const float b0  = b0p[0];
    const float b1  = b1p[0];
    const float one_m_ol = 1.0f - ol;

    float* __restrict__ h_n  = out;
    float* __restrict__ c_n  = out + (size_t)Bn;
    float* __restrict__ l_n  = out + 2 * (size_t)Bn;
    float* __restrict__ Goo  = out + 5 * (size_t)Bn;
    float* __restrict__ Gf   = out + 7 * (size_t)Bn;
    float* __restrict__ hout = out + 8 * (size_t)Bn;

    // Warm-up: contraction washes out the unknown incoming carry.
    float c = 0.0f;
    long long w0 = start - WARM;
    if (w0 < 0) w0 = 0;
    for (long long i = w0; i < start; ++i) {
        const float u  = x[(size_t)(tl + i) * 2];
        const float oo = oo1 * sigmoidf(b0 + (c - mo) / so * b1);
        const float f  = one_m_ol - oo;
        c = f * c + u;
    }

    // Main chunk: record pre-update values, then advance the carry.
    for (long long i = start; i < end; ++i) {
        const size_t row = (size_t)(tl + i);
        {   // keep the x stream ahead of the latency-bound chain
            size_t prow = row + 32;
            if (prow >= (size_t)Bn) prow = (size_t)Bn - 1;
            __builtin_prefetch(&x[prow * 2], 0, 1);
        }
        const float u  = x[row * 2];
        const float oo = oo1 * sigmoidf(b0 + (c - mo) / so * b1);
        const float f  = one_m_ol - oo;
        const float h  = oo * c;
        h_n[row]      = h;
        c_n[row]      = c;
        l_n[row]      = ol * c;
        Goo[row]      = oo;
        Gf[row]       = f;
        hout[2 * row] = h;
        c = f * c + u;
    }
}

// ---------------------------------------------------------------------------
// Kernel 2: unbiased std via WMMA reductions on the matrix pipe.
// One block, 32 waves. Each wave consumes 64 elements per WMMA pair.
// With A == ones(16x4): sum over ALL entries of D == 16 * sum(B elements),
// independent of the exact lane/VGPR layout -> layout-robust reduction.
// ---------------------------------------------------------------------------
__global__ void std_wmma_kernel(const float* __restrict__ y, int Bn,
                                float* __restrict__ ws) {
    const int n0 = SPIN_LEN;
    int n1 = TRAINTIME_LEN;
    if (n1 > Bn) n1 = Bn;
    const int n = n1 - n0;                 // 399000 for the given shapes
    const int wave   = threadIdx.x >> 5;
    const int lane   = threadIdx.x & 31;
    const int nwaves = blockDim.x >> 5;
    const int tiles  = (n + 63) >> 6;

    v8f accS = {};
    v8f accQ = {};
    v2f ones;
    ones.x = 1.0f;
    ones.y = 1.0f;

    for (int t = wave; t < tiles; t += nwaves) {   // uniform per wave: EXEC all-1
        const int i0 = n0 + t * 64 + lane * 2;
        const int i1 = i0 + 1;
        float y0 = y[i0 < n1 ? i0 : (n1 - 1)];
        float y1 = y[i1 < n1 ? i1 : (n1 - 1)];
        y0 = (i0 < n1) ? y0 : 0.0f;                // OOB contributes 0 to sums
        y1 = (i1 < n1) ? y1 : 0.0f;
        v2f bv; bv.x = y0;      bv.y = y1;
        v2f bq; bq.x = y0 * y0; bq.y = y1 * y1;
        accS = __builtin_amdgcn_wmma_f32_16x16x4_f32(
            false, ones, false, bv, (short)0, accS, false, false);
        accQ = __builtin_amdgcn_wmma_f32_16x16x4_f32(
            false, ones, false, bq, (short)0, accQ, false, false);
    }

    float ps = accS[0] + accS[1] + accS[2] + accS[3] +
               accS[4] + accS[5] + accS[6] + accS[7];
    float pq = accQ[0] + accQ[1] + accQ[2] + accQ[3] +
               accQ[4] + accQ[5] + accQ[6] + accQ[7];

    __shared__ float ls[1024];
    __shared__ float lq[1024];
    ls[threadIdx.x] = ps;
    lq[threadIdx.x] = pq;
    __syncthreads();
    for (int s = blockDim.x >> 1; s > 0; s >>= 1) {
        if ((int)threadIdx.x < s) {
            ls[threadIdx.x] += ls[threadIdx.x + s];
            lq[threadIdx.x] += lq[threadIdx.x + s];
        }
        __syncthreads();
    }
    if (threadIdx.x == 0) {
        const float S  = ls[0] * (1.0f / 16.0f);   // every element counted 16x
        const float Q  = lq[0] * (1.0f / 16.0f);
        const float nn = (float)n;
        const float var = (Q - S * S / nn) / (nn - 1.0f);
        ws[0] = sqrtf(fmaxf(var, 0.0f));
    }
}

// ---------------------------------------------------------------------------
// Kernel 3: constants, std broadcast, and time_lag zero padding.
// ---------------------------------------------------------------------------
__global__ void fill_kernel(const int*   __restrict__ time_lag_p,
                            const float* __restrict__ wo,
                            const float* __restrict__ wl,
                            const float* __restrict__ wf,
                            const float* __restrict__ ws,
                            float* __restrict__ out, int Bn) {
    const int b = blockIdx.x * blockDim.x + threadIdx.x;
    if (b >= Bn) return;
    const int tl = time_lag_p[0];
    const float eo = __expf(wo[0]);
    const float el = __expf(wl[0]);
    const float ef = __expf(wf[0]);
    const float ol = el / (eo + el + ef);
    const float sd = ws[0];
    const size_t B = (size_t)Bn;
    const bool pad = (b < tl);

    out[3 * B + b]         = 0.0f;               // bp_n
    out[4 * B + b]         = 0.0f;               // Gate_ib
    out[6 * B + b]         = pad ? 0.0f : ol;    // Gate_ol
    out[10 * B + b]        = pad ? 0.0f : sd;    // obs_std
    out[8 * B + 2 * b + 1] = pad ? 0.0f : sd;    // h_nout[:,1]
    if (pad) {                                   // rows the scan never writes
        out[b]             = 0.0f;               // h_n
        out[B + b]         = 0.0f;               // c_n
        out[2 * B + b]     = 0.0f;               // l_n
        out[5 * B + b]     = 0.0f;               // Gate_oo
        out[7 * B + b]     = 0.0f;               // Gate_f
        out[8 * B + 2 * b] = 0.0f;               // h_nout[:,0]
    }
}

// ---------------------------------------------------------------------------
extern "C" void kernel_launch(void* const* d_in, const int* in_sizes, int n_in,
                              void* d_out, int out_size, void* d_ws, size_t ws_size,
                              hipStream_t stream) {
    const float* x        = (const float*)d_in[0];
    // d_in[1] = epoch (unused by the math)
    const int*   time_lag = (const int*)d_in[2];
    const float* y_obs    = (const float*)d_in[3];
    const float* c_mean   = (const float*)d_in[4];
    const float* c_std    = (const float*)d_in[5];
    const float* wo       = (const float*)d_in[6];
    const float* wl       = (const float*)d_in[7];
    const float* wf       = (const float*)d_in[8];
    const float* b0       = (const float*)d_in[9];
    const float* b1       = (const float*)d_in[10];
    float*       out      = (float*)d_out;
    float*       ws       = (float*)d_ws;

    const int Bn = in_sizes[3];                  // y_obs has B rows

    const int scan_threads = (Bn + CHUNK - 1) / CHUNK;   // 512 for B=524288
    const int scan_blocks  = (scan_threads + 255) / 256;
    mcpbrnn_scan_kernel<<<scan_blocks, 256, 0, stream>>>(
        x, time_lag, c_mean, c_std, wo, wl, wf, b0, b1, out, Bn);

    std_wmma_kernel<<<1, 1024, 0, stream>>>(y_obs, Bn, ws);

    fill_kernel<<<(Bn + 255) / 256, 256, 0, stream>>>(
        time_lag, wo, wl, wf, ws, out, Bn);
}